// MOLGCirculantLinear_360777252981
// MI455X (gfx1250) — compile-verified
//
#include <hip/hip_runtime.h>

typedef __attribute__((ext_vector_type(2))) float v2f;
typedef __attribute__((ext_vector_type(8))) float v8f;

#define BATCH 512
#define GY    128
#define GXN   128
#define KCIRC 8
#define NCH   1024   // gy*k == gx*k == in_channel == out_channel

// One wave computes a 16(batch) x 16(out-channel) tile of the per-gx GEMM
//   xr[b, n] = sum_i W2[i, n] * x2[b, gx*8+i],  W2[i, gy*8+j] = w[gy,gx,(j-i)&7]
// using two chained native-f32 v_wmma_f32_16x16x4_f32 ops (K=0..3, K=4..7),
// looping gx = 0..127 and accumulating the photodetection sum in registers.
// Full f32 path: no precision loss vs the f32 reference.
__global__ __launch_bounds__(128)
void molg_circulant_wmma_kernel(const float* __restrict__ x,
                                const float* __restrict__ w,
                                const float* __restrict__ scale,
                                float* __restrict__ out,
                                float* __restrict__ phase)
{
    const int lane  = threadIdx.x & 31;
    const int wave  = threadIdx.x >> 5;
    const int ntile = blockIdx.x * 4 + wave;   // 0..63  (16 out-channels each)
    const int btile = blockIdx.y;              // 0..31  (16 batch rows each)

    const int laneN  = lane & 15;
    const int laneHi = lane >> 4;              // 32-bit A/B: lanes 16-31 carry K+2
    const int n      = ntile * 16 + laneN;     // output channel 0..1023
    const int gy     = n >> 3;
    const int j      = n & 7;
    const int brow   = btile * 16 + laneN;     // A-matrix row (batch index)
    const int k0     = laneHi * 2;             // this lane's K base within each WMMA

    // circulant rotation indices for this lane's B entries (compile-time-ish)
    const int jb0 = (j - k0) & 7;
    const int jb1 = (j - k0 - 1) & 7;
    const int jb2 = (j - k0 - 4) & 7;
    const int jb3 = (j - k0 - 5) & 7;

    float sum[8];
    #pragma unroll
    for (int r = 0; r < 8; ++r) sum[r] = 0.0f;

    for (int gx = 0; gx < GXN; ++gx) {
        // ---- A fragments: x^2, 16x4 f32 per WMMA; lane holds K=k0,k0+1 (+4) ----
        const float* xp = x + brow * NCH + gx * KCIRC;
        float xa0 = xp[k0];
        float xa1 = xp[k0 + 1];
        float xa2 = xp[k0 + 4];
        float xa3 = xp[k0 + 5];
        v2f a1, a2;
        a1[0] = xa0 * xa0;
        a1[1] = xa1 * xa1;
        a2[0] = xa2 * xa2;
        a2[1] = xa3 * xa3;

        // ---- B fragments: circulant-expanded weight column for n = gy*8+j ----
        const float* wp = w + (gy * GXN + gx) * KCIRC;
        v2f b1, b2;
        b1[0] = wp[jb0];
        b1[1] = wp[jb1];
        b2[0] = wp[jb2];
        b2[1] = wp[jb3];

        // hint upcoming weight line (gfx1250 global_prefetch_b8)
        if (gx + 4 < GXN)
            __builtin_prefetch(w + (gy * GXN + gx + 4) * KCIRC, 0, 3);

        // ---- D = A0*B0 + A1*B1 + 0 : two chained v_wmma_f32_16x16x4_f32 ----
        v8f c = {};
        c = __builtin_amdgcn_wmma_f32_16x16x4_f32(
                /*neg_a=*/false, a1, /*neg_b=*/false, b1,
                /*c_mod=*/(short)0, c, /*reuse_a=*/false, /*reuse_b=*/false);
        c = __builtin_amdgcn_wmma_f32_16x16x4_f32(
                /*neg_a=*/false, a2, /*neg_b=*/false, b2,
                /*c_mod=*/(short)0, c, /*reuse_a=*/false, /*reuse_b=*/false);

        const float s = scale[(gx & 63) * KCIRC + j];

        #pragma unroll
        for (int r = 0; r < 8; ++r) {
            const int   brw = btile * 16 + laneHi * 8 + r;   // batch row of this acc VGPR
            const float xr  = c[r];
            // phase output: [B, gy, gx, k]
            phase[(((brw * GY) + gy) * GXN + gx) * KCIRC + j] = xr;
            // degree-5 polynomial, highest coeff first
            float p = 0.02403f;
            p = p * xr + (-0.2699f);
            p = p * xr + 1.17f;
            p = p * xr + (-2.454f);
            p = p * xr + 2.523f;
            p = p * xr + (-0.08003f);
            p = fminf(fmaxf(p, 0.0f), 1.0f);
            sum[r] += p * s;
        }
    }

    // out[b, n] = sum_gx (2*xm - 1) = 2*sum - 128
    #pragma unroll
    for (int r = 0; r < 8; ++r) {
        const int brw = btile * 16 + laneHi * 8 + r;
        out[brw * NCH + n] = 2.0f * sum[r] - (float)GXN;
    }
}

extern "C" void kernel_launch(void* const* d_in, const int* in_sizes, int n_in,
                              void* d_out, int out_size, void* d_ws, size_t ws_size,
                              hipStream_t stream) {
    const float* x     = (const float*)d_in[0];   // [512, 1024]
    const float* wgt   = (const float*)d_in[1];   // [128, 128, 8]
    const float* scale = (const float*)d_in[2];   // [1, 1, 64, 8]

    float* out   = (float*)d_out;                 // [512, 1024]
    float* phase = out + BATCH * NCH;             // [512, 128, 128, 8]

    dim3 grid(16, 32);    // 16 ntile-groups (x4 waves), 32 btiles
    dim3 block(128);      // 4 waves (wave32)
    molg_circulant_wmma_kernel<<<grid, block, 0, stream>>>(x, wgt, scale, out, phase);
}